// AutoEncoder_53747220742606
// MI455X (gfx1250) — compile-verified
//
#include <hip/hip_runtime.h>
#include <math.h>

typedef float v2f __attribute__((ext_vector_type(2)));
typedef float v8f __attribute__((ext_vector_type(8)));

#define BATCH   131072
#define MC      50
#define ESIZE   10
#define DSIZE   20
#define LSTRIDE 672   // floats per prepped layer: 2 coltiles * 5 ksteps * 64 + 32 (t-vector)
#define TOFF    640   // offset of t-vector within a layer slab
#define ASTRIDE 33    // LDS activation row stride in floats (padded from 32 -> fewer bank conflicts)

// ---------------------------------------------------------------------------
// Prep: fold BN into weights, pad, and swizzle into the WMMA B-operand layout.
// B-matrix (4x16 f32) layout assumption (mirror of documented A layout):
//   lane l holds column n = l%16, K-pair k0 = 2*(l/16) within each K=4 slice.
// We store, per layer: [coltile][kstep][lane][2] floats, then t[32].
// ---------------------------------------------------------------------------
__global__ void prep_layer_kernel(const float* __restrict__ w,
                                  const float* __restrict__ b,
                                  const float* __restrict__ gamma,
                                  const float* __restrict__ beta,
                                  const float* __restrict__ rm,
                                  const float* __restrict__ rv,
                                  int fin, int fout, int hidden,
                                  float* __restrict__ dst) {
  const int lane = threadIdx.x;  // 32 threads
  // t-vector (bias + BN shift), zero for padded columns
  {
    const int n = lane;
    float t = 0.f;
    if (n < fout) {
      if (hidden) {
        const float s = gamma[n] * rsqrtf(rv[n] + 1e-5f);
        t = (b[n] - rm[n]) * s + beta[n];
      } else {
        t = b[n];
      }
    }
    dst[TOFF + lane] = t;
  }
  // swizzled, BN-scaled weights
  for (int c = 0; c < 2; ++c) {
    const int n = c * 16 + (lane & 15);
    float s = 0.f;
    if (n < fout) s = hidden ? gamma[n] * rsqrtf(rv[n] + 1e-5f) : 1.f;
    for (int st = 0; st < 5; ++st) {
      const int k0 = 4 * st + 2 * (lane >> 4);
      float w0 = 0.f, w1 = 0.f;
      if (n < fout) {
        if (k0 < fin)     w0 = w[(size_t)k0 * fout + n] * s;
        if (k0 + 1 < fin) w1 = w[(size_t)(k0 + 1) * fout + n] * s;
      }
      float* p = dst + ((c * 5 + st) * 64 + lane * 2);
      p[0] = w0;
      p[1] = w1;
    }
  }
}

// ---------------------------------------------------------------------------
// One 16x16 output tile: chain of f32 16x16x4 WMMAs over the K dimension.
// actIn: 16 rows x ASTRIDE floats in LDS. A layout: lane m=l%16, kpair 2*(l/16).
// ---------------------------------------------------------------------------
template <int KSTEPS>
__device__ __forceinline__ v8f mm_tile(const float* __restrict__ actIn,
                                       const float* __restrict__ layerWs,
                                       int lane, int ctile) {
  v8f acc = {0.f, 0.f, 0.f, 0.f, 0.f, 0.f, 0.f, 0.f};
  const int m = lane & 15;
  const int koff = (lane >> 4) << 1;
  const float* wp = layerWs + (ctile * 5) * 64 + lane * 2;
#pragma unroll
  for (int s = 0; s < KSTEPS; ++s) {
    v2f a;
    a.x = actIn[m * ASTRIDE + 4 * s + koff];
    a.y = actIn[m * ASTRIDE + 4 * s + koff + 1];
    v2f bb;
    bb.x = wp[0];
    bb.y = wp[1];
    wp += 64;
    acc = __builtin_amdgcn_wmma_f32_16x16x4_f32(false, a, false, bb,
                                                (short)0, acc, false, false);
  }
  return acc;
}

// Hidden layer: WMMA tiles + (folded-BN bias) + ReLU -> LDS (C/D layout store).
template <int KSTEPS, int NCOLS>
__device__ __forceinline__ void hidden_layer(const float* __restrict__ actIn,
                                             float* __restrict__ actOut,
                                             const float* __restrict__ layerWs,
                                             int lane) {
#pragma unroll
  for (int c = 0; c < NCOLS; ++c) {
    v8f acc = mm_tile<KSTEPS>(actIn, layerWs, lane, c);
    const int n = c * 16 + (lane & 15);
    const float t = layerWs[TOFF + n];
    const int rbase = (lane >> 4) * 8;
#pragma unroll
    for (int r = 0; r < 8; ++r) {
      actOut[(rbase + r) * ASTRIDE + n] = fmaxf(acc[r] + t, 0.f);
    }
  }
}

// Final layer. MODE: 0 = identity (em), 1 = 0.05+sigmoid (ev), 2 = sigmoid (x_pred)
template <int KSTEPS, int NCOLS, int MODE>
__device__ __forceinline__ void final_layer(const float* __restrict__ actIn,
                                            const float* __restrict__ layerWs,
                                            int lane, int fout,
                                            float* __restrict__ outp,
                                            size_t rowbase, int ldo) {
#pragma unroll
  for (int c = 0; c < NCOLS; ++c) {
    v8f acc = mm_tile<KSTEPS>(actIn, layerWs, lane, c);
    const int n = c * 16 + (lane & 15);
    const float t = layerWs[TOFF + n];
    const int rbase = (lane >> 4) * 8;
    if (n < fout) {
#pragma unroll
      for (int r = 0; r < 8; ++r) {
        float v = acc[r] + t;
        if (MODE != 0) {
          v = 1.f / (1.f + __expf(-v));
          if (MODE == 1) v += 0.05f;
        }
        outp[(rowbase + (size_t)(rbase + r)) * (size_t)ldo + n] = v;
      }
    }
  }
}

// ---------------------------------------------------------------------------
// Encoder: both MLPs (mean + var). 8 waves/block, 16 rows/wave.
// ENC dims 20->20->16->14->12->10  =>  ksteps 5,5,4,4,3 ; ncols 2,1,1,1,1
// ---------------------------------------------------------------------------
__global__ void __launch_bounds__(256) enc_kernel(const float* __restrict__ x,
                                                  const float* __restrict__ ws,
                                                  float* __restrict__ em,
                                                  float* __restrict__ ev) {
  __shared__ float lds[8 * 2 * 16 * ASTRIDE];
  const int lane = threadIdx.x & 31;
  const int wv = threadIdx.x >> 5;
  float* b0 = lds + wv * (2 * 16 * ASTRIDE);
  float* b1 = b0 + 16 * ASTRIDE;
  const size_t rowbase = ((size_t)blockIdx.x * 8 + wv) * 16;

  // load x tile (pad cols >= 20 with zero)
  for (int idx = lane; idx < 512; idx += 32) {
    const int r = idx >> 5, c = idx & 31;
    b0[r * ASTRIDE + c] = (c < DSIZE) ? x[(rowbase + r) * DSIZE + c] : 0.f;
  }

  // enc_mean (ws layers 0..4)
  hidden_layer<5, 2>(b0, b1, ws + 0 * LSTRIDE, lane);
  hidden_layer<5, 1>(b1, b0, ws + 1 * LSTRIDE, lane);
  hidden_layer<4, 1>(b0, b1, ws + 2 * LSTRIDE, lane);
  hidden_layer<4, 1>(b1, b0, ws + 3 * LSTRIDE, lane);
  final_layer<3, 1, 0>(b0, ws + 4 * LSTRIDE, lane, ESIZE, em, rowbase, ESIZE);

  // reload x (b0 was clobbered; x tile is L2-hot)
  for (int idx = lane; idx < 512; idx += 32) {
    const int r = idx >> 5, c = idx & 31;
    b0[r * ASTRIDE + c] = (c < DSIZE) ? x[(rowbase + r) * DSIZE + c] : 0.f;
  }

  // enc_var (ws layers 5..9), ev = 0.05 + sigmoid(.)
  hidden_layer<5, 2>(b0, b1, ws + 5 * LSTRIDE, lane);
  hidden_layer<5, 1>(b1, b0, ws + 6 * LSTRIDE, lane);
  hidden_layer<4, 1>(b0, b1, ws + 7 * LSTRIDE, lane);
  hidden_layer<4, 1>(b1, b0, ws + 8 * LSTRIDE, lane);
  final_layer<3, 1, 1>(b0, ws + 9 * LSTRIDE, lane, ESIZE, ev, rowbase, ESIZE);
}

// ---------------------------------------------------------------------------
// Decoder: fuses z = em + sqrt(ev)*eps with the 5-layer decoder MLP.
// DEC dims 10->12->14->16->20->20 => ksteps 3,3,4,4,5 ; ncols 1,1,1,2,2
// ---------------------------------------------------------------------------
__global__ void __launch_bounds__(256) dec_kernel(const float* __restrict__ ws,
                                                  const float* __restrict__ em,
                                                  const float* __restrict__ ev,
                                                  const float* __restrict__ eps,
                                                  float* __restrict__ z,
                                                  float* __restrict__ xp) {
  __shared__ float lds[8 * 2 * 16 * ASTRIDE];
  const int lane = threadIdx.x & 31;
  const int wv = threadIdx.x >> 5;
  float* b0 = lds + wv * (2 * 16 * ASTRIDE);
  float* b1 = b0 + 16 * ASTRIDE;

  const size_t g0 = ((size_t)blockIdx.x * 8 + wv) * 16;  // row in [0, MC*BATCH)
  const size_t mc = g0 >> 17;                            // BATCH = 2^17
  const size_t rowB = g0 & (size_t)(BATCH - 1);

  // zero the K-pad columns 10,11 read by the first layer (K padded 10 -> 12)
  {
    const int r = lane & 15;
    const int c = 10 + (lane >> 4);
    b0[r * ASTRIDE + c] = 0.f;
  }
  // z tile: coalesced em/ev/eps reads, coalesced z write, staged into LDS
  for (int idx = lane; idx < 16 * ESIZE; idx += 32) {
    const int r = idx / ESIZE, e = idx % ESIZE;
    const size_t ri = rowB + (size_t)r;
    const float m = em[ri * ESIZE + e];
    const float v = ev[ri * ESIZE + e];
    const float ep = eps[(mc * BATCH + ri) * ESIZE + e];
    const float zz = m + sqrtf(v) * ep;
    z[g0 * ESIZE + idx] = zz;
    b0[r * ASTRIDE + e] = zz;
  }

  hidden_layer<3, 1>(b0, b1, ws + 10 * LSTRIDE, lane);
  hidden_layer<3, 1>(b1, b0, ws + 11 * LSTRIDE, lane);
  hidden_layer<4, 1>(b0, b1, ws + 12 * LSTRIDE, lane);
  hidden_layer<4, 2>(b1, b0, ws + 13 * LSTRIDE, lane);
  final_layer<5, 2, 2>(b0, ws + 14 * LSTRIDE, lane, DSIZE, xp, g0, DSIZE);
}

// ---------------------------------------------------------------------------
// Host launcher
// d_in flatten order: x; enc_mean L0..L4; enc_var L0..L4; dec_mean L0..L4; eps.
// Hidden layer dict keys sorted: b, beta, gamma, rm, rv, w ; final: b, w.
// ---------------------------------------------------------------------------
extern "C" void kernel_launch(void* const* d_in, const int* in_sizes, int n_in,
                              void* d_out, int out_size, void* d_ws, size_t ws_size,
                              hipStream_t stream) {
  const float* x = (const float*)d_in[0];
  float* ws = (float*)d_ws;

  static const int ENCD[6] = {20, 20, 16, 14, 12, 10};
  static const int DECD[6] = {10, 12, 14, 16, 20, 20};

  int idx = 1;
  auto prep_mlp = [&](const int* dims, int base) {
    for (int L = 0; L < 5; ++L) {
      const int fin = dims[L], fout = dims[L + 1];
      const int hidden = (L < 4) ? 1 : 0;
      const float *b, *beta, *gamma, *rm, *rv, *w;
      if (hidden) {
        b     = (const float*)d_in[idx + 0];
        beta  = (const float*)d_in[idx + 1];
        gamma = (const float*)d_in[idx + 2];
        rm    = (const float*)d_in[idx + 3];
        rv    = (const float*)d_in[idx + 4];
        w     = (const float*)d_in[idx + 5];
        idx += 6;
      } else {
        b = (const float*)d_in[idx + 0];
        w = (const float*)d_in[idx + 1];
        idx += 2;
        beta = gamma = rm = rv = b;  // unused when hidden==0
      }
      prep_layer_kernel<<<dim3(1), dim3(32), 0, stream>>>(
          w, b, gamma, beta, rm, rv, fin, fout, hidden,
          ws + (size_t)(base + L) * LSTRIDE);
    }
  };
  prep_mlp(ENCD, 0);   // enc_mean -> ws layers 0..4
  prep_mlp(ENCD, 5);   // enc_var  -> ws layers 5..9
  prep_mlp(DECD, 10);  // dec_mean -> ws layers 10..14

  const float* eps = (const float*)d_in[n_in - 1];

  float* em = (float*)d_out;
  float* ev = em + (size_t)BATCH * ESIZE;
  float* z  = ev + (size_t)BATCH * ESIZE;
  float* xp = z + (size_t)MC * BATCH * ESIZE;

  enc_kernel<<<dim3(BATCH / 128), dim3(256), 0, stream>>>(x, ws, em, ev);
  dec_kernel<<<dim3((MC * BATCH) / 128), dim3(256), 0, stream>>>(ws, em, ev, eps, z, xp);
}